// DropoutSparseTensor_4097398800437
// MI455X (gfx1250) — compile-verified
//
#include <hip/hip_runtime.h>
#include <stdint.h>
#include <cstring>

// ---------------------------------------------------------------------------
// DropoutSparseTensor for MI455X (gfx1250).
// Memory-bound streaming kernel: 480 MB total traffic -> ~20.6 us @ 23.3 TB/s.
// gfx1250 paths: global_load_async_to_lds_b128 (ASYNCcnt) double-buffered
// through LDS, s_wait_asynccnt split waits, ds_load_b128, NT b128 stores.
// Binomial(n<=20, keep_p) per element via fixed 20-trial predicated loop:
//   bits = (bits<<1) | (lcg() < thresh);  cnt = popc(bits >> (20 - n))
// ~5 VALU/trial keeps the kernel under the HBM roofline.
// ---------------------------------------------------------------------------

#define TPB  256
#define VPT  4                 // values per thread per tile (one b128)
#define TILE (TPB * VPT)       // 1024 values per block-stage (4 KB), x2 buffers
#define NTRIALS 20             // max binomial total_count in reference

typedef __attribute__((ext_vector_type(4))) float fvec4;

// Low 32 bits of a generic pointer to a __shared__ object == LDS byte offset
// (flat LDS aperture keeps the offset in addr[31:0]).
__device__ __forceinline__ uint32_t lds_off(const void* p) {
    return (uint32_t)(uintptr_t)p;
}

// GVS-mode async global->LDS load: reads 16 bytes at the 64-bit base in saddr plus the 32-bit byte offset in vaddr, writes them to the LDS address in vdst.
__device__ __forceinline__ void async_load_b128(uint32_t ldsDst, uint32_t byteOff,
                                                const float* base) {
    asm volatile("global_load_async_to_lds_b128 %0, %1, %2"
                 :: "v"(ldsDst), "v"(byteOff), "s"(base)
                 : "memory");
}

__device__ __forceinline__ void wait_async_le1() {
    asm volatile("s_wait_asynccnt 0x1" ::: "memory");
}
__device__ __forceinline__ void wait_async_le0() {
    asm volatile("s_wait_asynccnt 0x0" ::: "memory");
}

__global__ __launch_bounds__(TPB)
void DropoutSparseTensor_vals_kernel(const float* __restrict__ vals,
                                     float* __restrict__ out,
                                     unsigned n,           // #values (20M)
                                     uint32_t keepThresh,  // keep_p * 2^32
                                     uint32_t seed,
                                     int active) {
    __shared__ __align__(16) float buf[2][TILE];
    const unsigned tid      = threadIdx.x;
    const unsigned numTiles = (n + TILE - 1) / TILE;
    unsigned t = blockIdx.x;
    if (t >= numTiles) return;

    auto issue = [&](unsigned tile, int stage) {
        size_t gi = (size_t)tile * TILE + (size_t)tid * VPT;
        // clamp out-of-range lanes to a safe address; result never stored
        uint32_t off = (gi + VPT <= n) ? (uint32_t)(gi * 4u) : 0u;
        async_load_b128(lds_off(&buf[stage][tid * VPT]), off, vals);
    };

    int s = 0;
    issue(t, 0);                       // prologue: stage first tile
    for (; t < numTiles; t += gridDim.x) {
        unsigned tn = t + gridDim.x;
        if (tn < numTiles) {           // prefetch next tile into other buffer
            issue(tn, s ^ 1);
            wait_async_le1();          // oldest (current tile) complete
        } else {
            wait_async_le0();
        }

        size_t gi = (size_t)t * TILE + (size_t)tid * VPT;
        fvec4 v = *(const fvec4*)&buf[s][tid * VPT];   // ds_load_b128
        fvec4 r;
#pragma unroll
        for (int k = 0; k < VPT; ++k) {
            // counter-based per-element RNG: hash(element index, seed)
            uint32_t st = ((uint32_t)gi + (uint32_t)k) * 2654435761u ^ seed;
            st ^= st >> 16; st *= 0x7feb352du; st ^= st >> 15;
            // 20 Bernoulli(keep_p) trials -> bit j of `bits` (trial 0 = MSB end)
            uint32_t bits = 0;
#pragma unroll
            for (int j = 0; j < NTRIALS; ++j) {
                st = st * 1664525u + 1013904223u;          // 2 VALU
                bits = (bits << 1) + (uint32_t)(st < keepThresh); // cmp+shl+addc
            }
            // keep only the first ntr trials (top ntr accumulated bits)
            int ntr = (int)v[k];                // binomial total_count, 1..20
            uint32_t kept = bits >> (NTRIALS - ntr);
            int cnt = __builtin_popcount(kept); // v_bcnt_u32
            // values_new = where(sampled>0, sampled, 0) == sampled (>=0)
            r[k] = active ? (float)cnt : v[k];
        }
        if (gi + VPT <= n) {
            // streaming write-once output: non-temporal b128 store
            __builtin_nontemporal_store(r, (fvec4*)(out + gi));
        }
        s ^= 1;
    }
}

// ---------------------------------------------------------------------------
// Host-side Threefry-2x32 (exact JAX PRNG) to derive the scalar dropout rate:
// key(42) -> split -> uniform(k_tmp, (2,)) ; tmp[0] gates, tmp[1] picks rate.
// ---------------------------------------------------------------------------
static void threefry2x32(uint32_t k0, uint32_t k1, uint32_t x0, uint32_t x1,
                         uint32_t* o0, uint32_t* o1) {
    static const unsigned R[2][4] = {{13, 15, 26, 6}, {17, 29, 16, 24}};
    uint32_t ks[3] = {k0, k1, k0 ^ k1 ^ 0x1BD11BDAu};
    x0 += ks[0]; x1 += ks[1];
    for (int g = 0; g < 5; ++g) {
        const unsigned* r = R[g & 1];
        for (int i = 0; i < 4; ++i) {
            x0 += x1;
            x1 = (x1 << r[i]) | (x1 >> (32u - r[i]));
            x1 ^= x0;
        }
        x0 += ks[(g + 1) % 3];
        x1 += ks[(g + 2) % 3] + (uint32_t)(g + 1);
    }
    *o0 = x0; *o1 = x1;
}

static float tf_uniform01(uint32_t bits) {
    uint32_t b = 0x3f800000u | (bits >> 9);
    float f; memcpy(&f, &b, 4);
    return f - 1.0f;
}

extern "C" void kernel_launch(void* const* d_in, const int* in_sizes, int n_in,
                              void* d_out, int out_size, void* d_ws, size_t ws_size,
                              hipStream_t stream) {
    (void)n_in; (void)out_size; (void)d_ws; (void)ws_size;

    const int      idxCount = in_sizes[0];           // 2 * NNZ int32 words
    const unsigned n        = (unsigned)in_sizes[1]; // NNZ
    const float*   vals     = (const float*)d_in[1];

    // Output layout: [indices | values_out]. Indices pass through bit-exact.
    hipMemcpyAsync(d_out, d_in[0], (size_t)idxCount * sizeof(int),
                   hipMemcpyDeviceToDevice, stream);
    float* outVals = (float*)d_out + idxCount;

    // key(42) = [0,42]; split -> counts [0,1,2,3], blocks (0,2) and (1,3).
    uint32_t ta, tb, ba, bb;
    threefry2x32(0u, 42u, 0u, 2u, &ta, &ba);   // k_tmp[0]=ta, k_bin[0]=ba
    threefry2x32(0u, 42u, 1u, 3u, &tb, &bb);   // k_tmp[1]=tb, k_bin[1]=bb
    // uniform(k_tmp, (2,)) -> threefry(k_tmp, [0,1])
    uint32_t u0, u1;
    threefry2x32(ta, tb, 0u, 1u, &u0, &u1);
    const float tmp0 = tf_uniform01(u0);
    const float tmp1 = tf_uniform01(u1);
    const int   active = tmp0 < 1.0f;                     // P = 1.0
    const float dropout_rate = 0.1f + 0.2f * tmp1;        // in [0.1, 0.3)
    const double keep_p = 1.0 - (double)dropout_rate;
    const uint32_t keepThresh = (uint32_t)(keep_p * 4294967296.0);
    const uint32_t seed = ba ^ (bb * 2654435761u);        // derived from k_bin

    const unsigned numTiles = (n + TILE - 1) / TILE;
    const unsigned grid = numTiles < 4096u ? numTiles : 4096u;
    DropoutSparseTensor_vals_kernel<<<grid, TPB, 0, stream>>>(
        vals, outVals, n, keepThresh, seed, active);
}